// LiquidLayer_1005022347993
// MI455X (gfx1250) — compile-verified
//
#include <hip/hip_runtime.h>

// ---------------------------------------------------------------------------
// LiquidLayer fused kernel for gfx1250 (MI455X, wave32, WMMA), v2
//
//   I = x @ W^T + b ;  u' = u + (I-u)*a ;  s' = s + (u-s)*a ;  out = tanh(s')
//   B=16, S=2048, FIN=FOUT=512.
//
// Block = 4 waves (128 threads) = one (batch, 32-channel tile).
//   * K=512 reduction split across the 4 waves (128 each) -> 1024 waves total.
//   * Each wave's W slice (2 N-tiles x 32 k-steps x v2f) preloaded into
//     128 VGPRs: W is read from HBM exactly once per block (16 MB total).
//   * x chunk (16 t x 512 k, 32 KB) staged in padded LDS, shared by all waves;
//     L2 (192 MB) holds all of x so cross-block rereads are cheap.
//   * Partial accumulators combined with ds_add_f32 into a padded scan buffer;
//     wave 0 runs the sequential 16-step recurrence with branch-free tanh.
// ---------------------------------------------------------------------------

typedef __attribute__((ext_vector_type(2))) float v2f;
typedef __attribute__((ext_vector_type(8))) float v8f;

#define B_   16
#define S_   2048
#define K_   512
#define O_   512
#define NW   4            // waves per block (K-split factor)
#define KW   (K_ / NW)    // K range per wave = 128
#define XLD  (K_ + 4)     // padded LDS row stride (dwords) -> conflict-free frags
#define SBLD 33           // padded scan-buffer row stride

__global__ __launch_bounds__(128)
void liquid_layer_kernel(const float* __restrict__ x,
                         const float* __restrict__ W,
                         const float* __restrict__ bias,
                         const float* __restrict__ tau,
                         float* __restrict__ out)
{
    __shared__ float XT[16 * XLD];   // x chunk: 16 timesteps x 512 k (padded)
    __shared__ float SB[16 * SBLD];  // combined I tile: 16 t x 32 n (padded)

    const int tid  = threadIdx.x;
    const int lane = tid & 31;           // lane within wave32
    const int w    = tid >> 5;           // wave id 0..3 (K-split index)
    const int b    = blockIdx.x >> 4;    // batch
    const int o0   = (blockIdx.x & 15) * 32;

    // WMMA 16x16x4 f32 fragment coords:
    //  A: lanes 0-15 -> M=lane, K={0,1}; lanes 16-31 -> M=lane-16, K={2,3}
    //  B: lanes 0-15 -> N=lane, K={0,1}; lanes 16-31 -> N=lane-16, K={2,3}
    const int mrow = lane & 15;
    const int kb   = (lane >> 4) * 2;
    const int k0   = w * KW;             // this wave's K range

    // ---- Preload this wave's W fragments into registers (once per block) ----
    const float* wptr0 = W + (size_t)(o0 + mrow) * K_ + k0 + kb;
    const float* wptr1 = W + (size_t)(o0 + 16 + mrow) * K_ + k0 + kb;
    v2f wf0[KW / 4], wf1[KW / 4];        // 2 x 32 x v2f = 128 VGPRs
    #pragma unroll
    for (int kk = 0; kk < KW / 4; ++kk) {
        wf0[kk] = *(const v2f*)(wptr0 + kk * 4);
        wf1[kk] = *(const v2f*)(wptr1 + kk * 4);
    }

    // ---- Scan state (meaningful in wave 0 only; one channel per lane) ----
    const int   ch     = o0 + lane;
    const float a_gain = 1.0f / tau[ch];  // dt = 1.0
    const float bchan  = bias[ch];
    float u = 0.0f, s = 0.0f;

    const float* xb = x + (size_t)b * S_ * K_;

    for (int t0 = 0; t0 < S_; t0 += 16) {
        __syncthreads();  // previous chunk's scan done before XT/SB reuse

        // ---- Stage x[b, t0:t0+16, :] into LDS; wave w loads rows 4w..4w+3 ----
        #pragma unroll
        for (int rr = 0; rr < 4; ++rr) {
            const int    row = w * 4 + rr;
            const float* src = xb + (size_t)(t0 + row) * K_;
            float*       dst = &XT[row * XLD];
            #pragma unroll
            for (int j = 0; j < 4; ++j) {
                const int c = (j * 32 + lane) * 4;
                *(float4*)(dst + c) = *(const float4*)(src + c);  // 512B/instr
            }
        }
        // Zero the scan buffer (512 entries / 128 threads = 4 each).
        #pragma unroll
        for (int i = 0; i < 4; ++i) {
            const int idx = tid + i * 128;
            SB[(idx >> 5) * SBLD + (idx & 31)] = 0.0f;
        }
        // Prefetch next chunk (128 threads x 256B = 32 KB).
        if (t0 + 16 < S_) {
            __builtin_prefetch(xb + (size_t)(t0 + 16) * K_ + tid * 64, 0, 1);
        }

        __syncthreads();  // XT staged, SB zeroed

        // ---- GEMM: this wave covers K in [k0, k0+128), fully unrolled ----
        v8f acc0 = {0.f,0.f,0.f,0.f,0.f,0.f,0.f,0.f};
        v8f acc1 = {0.f,0.f,0.f,0.f,0.f,0.f,0.f,0.f};
        const float* xf = &XT[mrow * XLD + k0 + kb];

        #pragma unroll
        for (int kk = 0; kk < KW / 4; ++kk) {
            v2f afrag = *(const v2f*)(xf + kk * 4);   // LDS, conflict-free
            acc0 = __builtin_amdgcn_wmma_f32_16x16x4_f32(
                       false, afrag, false, wf0[kk], (short)0, acc0, false, false);
            acc1 = __builtin_amdgcn_wmma_f32_16x16x4_f32(
                       false, afrag, false, wf1[kk], (short)0, acc1, false, false);
        }

        // ---- Combine K-split partials into SB[t][n] via ds_add_f32 ----
        // D layout: VGPR r, lanes 0-15 -> M=r, N=lane; lanes 16-31 -> M=r+8.
        {
            const int mbase = (lane >> 4) * 8;
            const int n     = lane & 15;
            #pragma unroll
            for (int r = 0; r < 8; ++r) {
                atomicAdd(&SB[(mbase + r) * SBLD + n],      acc0[r]);
                atomicAdd(&SB[(mbase + r) * SBLD + n + 16], acc1[r]);
            }
        }

        __syncthreads();  // all partials landed

        // ---- Wave 0: sequential 16-step scan, one channel per lane ----
        if (w == 0) {
            float* orow = out + ((size_t)b * S_ + t0) * O_ + ch;
            #pragma unroll
            for (int tt = 0; tt < 16; ++tt) {
                const float I  = SB[tt * SBLD + lane] + bchan;
                const float un = fmaf(I - u, a_gain, u);   // u + (I - u)*a
                const float sn = fmaf(u - s, a_gain, s);   // s + (u_old - s)*a
                // Branch-free tanh: 1 - 2/(e^{2x}+1); exact at +/-inf limits.
                const float e  = __expf(2.0f * sn);
                orow[(size_t)tt * O_] = 1.0f - 2.0f / (e + 1.0f);
                u = un; s = sn;
            }
        }
    }
}

extern "C" void kernel_launch(void* const* d_in, const int* in_sizes, int n_in,
                              void* d_out, int out_size, void* d_ws, size_t ws_size,
                              hipStream_t stream) {
    const float* x    = (const float*)d_in[0];
    const float* W    = (const float*)d_in[1];
    const float* bias = (const float*)d_in[2];
    const float* tau  = (const float*)d_in[3];
    float* out = (float*)d_out;

    dim3 grid(B_ * (O_ / 32));  // 16 batches x 16 channel-tiles = 256 blocks
    dim3 block(NW * 32);        // 4 waves (wave32) per block
    hipLaunchKernelGGL(liquid_layer_kernel, grid, block, 0, stream,
                       x, W, bias, tau, out);
}